// KipfNetDoublePool_78039555768471
// MI455X (gfx1250) — compile-verified
//
#include <hip/hip_runtime.h>
#include <hip/hip_bf16.h>

#define NN 200000
#define NE 800000
#define NC1 20000
#define NE1 80000
#define NC2 2000
#define NE2 8000
#define FIN 24
#define FHID 64
#define KCH 6
#define FMIX 216
#define FOUT 6
#define BN_EPS 1e-5f

typedef float v2f __attribute__((ext_vector_type(2)));
typedef float v8f __attribute__((ext_vector_type(8)));

// ---------------- graph norm ----------------
__global__ void deg_kernel(const int* __restrict__ dst, float* __restrict__ deg, int E) {
    int e = blockIdx.x * blockDim.x + threadIdx.x;
    if (e < E) atomicAdd(&deg[dst[e]], 1.0f);
}

__global__ void dis_kernel(float* __restrict__ deg, int n) {
    int i = blockIdx.x * blockDim.x + threadIdx.x;
    if (i < n) {
        float d = deg[i];
        deg[i] = (d > 0.0f) ? rsqrtf(d) : 0.0f;   // deg counts are >=1 when >0
    }
}

__global__ void norm_kernel(const int* __restrict__ src, const int* __restrict__ dst,
                            const float* __restrict__ dis, float* __restrict__ norm, int E) {
    int e = blockIdx.x * blockDim.x + threadIdx.x;
    if (e < E) norm[e] = -dis[src[e]] * dis[dst[e]];
}

// ---------------- pooling (segment mean) ----------------
__global__ void pool_scatter(const float* __restrict__ x, const int* __restrict__ cl,
                             float* __restrict__ sum, float* __restrict__ cnt, int n) {
    int i = blockIdx.x * blockDim.x + threadIdx.x;
    if (i >= n) return;
    int c = cl[i];
    atomicAdd(&cnt[c], 1.0f);
    const float* xr = x + (size_t)i * FIN;
    float* sr = sum + (size_t)c * FIN;
    #pragma unroll
    for (int f = 0; f < FIN; ++f) atomicAdd(&sr[f], xr[f]);
}

__global__ void pool_div(float* __restrict__ sum, const float* __restrict__ cnt, int C) {
    int i = blockIdx.x * blockDim.x + threadIdx.x;
    if (i < C * FIN) {
        float c = cnt[i / FIN];
        sum[i] = sum[i] / fmaxf(c, 1.0f);
    }
}

// ---------------- Chebyshev propagation (scatter) ----------------
// Tout[dst,:] += alpha * norm[e] * Tsrc[src,:]
__global__ void scatter_prop(const float* __restrict__ Tsrc, const int* __restrict__ src,
                             const int* __restrict__ dst, const float* __restrict__ norm,
                             float alpha, float* __restrict__ Tout, int E) {
    int e = blockIdx.x * blockDim.x + threadIdx.x;
    if (e >= E) return;
    float nv = alpha * norm[e];
    const float* sr = Tsrc + (size_t)src[e] * FIN;
    float* dr = Tout + (size_t)dst[e] * FIN;
    #pragma unroll
    for (int f = 0; f < FIN; ++f) atomicAdd(&dr[f], nv * sr[f]);
}

// out = -a   (init for Tx2 = 2*prop(Tx1) - Tx0 recurrence)
__global__ void neg_copy(const float* __restrict__ a, float* __restrict__ o, int n) {
    int i = blockIdx.x * blockDim.x + threadIdx.x;
    if (i < n) o[i] = -a[i];
}

// ---------------- WMMA: out(n x 64) += T(n x 24) @ W(24 x 64), f32 ----------------
// One wave per 16-row tile; 4 column tiles of 16; K = 24 in 6 steps of 4.
// Uses native V_WMMA_F32_16X16X4_F32. n must be a multiple of 16 (all our n are).
__global__ void wmma_acc_kernel(const float* __restrict__ T, const float* __restrict__ W,
                                float* __restrict__ out, int ntiles) {
    int wid = blockIdx.x * (blockDim.x >> 5) + (threadIdx.x >> 5);
    if (wid >= ntiles) return;                      // wave-uniform: EXEC stays all-ones
    int lane = threadIdx.x & 31;
    int lm = lane & 15;                             // row/col within tile
    int hk = (lane >> 4) << 1;                      // K sub-offset: 0 or 2
    int hr = (lane >> 4) << 3;                      // C row offset: 0 or 8
    int m0 = wid << 4;
    const float* trow = T + (size_t)(m0 + lm) * FIN;

    v8f c[4];
    #pragma unroll
    for (int j = 0; j < 4; ++j)
        #pragma unroll
        for (int v = 0; v < 8; ++v)
            c[j][v] = out[(size_t)(m0 + hr + v) * FHID + j * 16 + lm];

    #pragma unroll
    for (int k = 0; k < FIN; k += 4) {
        v2f a;
        a.x = trow[k + hk];
        a.y = trow[k + hk + 1];
        #pragma unroll
        for (int j = 0; j < 4; ++j) {
            v2f b;
            b.x = W[(k + hk) * FHID + j * 16 + lm];
            b.y = W[(k + hk + 1) * FHID + j * 16 + lm];
            c[j] = __builtin_amdgcn_wmma_f32_16x16x4_f32(
                false, a, false, b, (short)0, c[j], false, false);
        }
    }

    #pragma unroll
    for (int j = 0; j < 4; ++j)
        #pragma unroll
        for (int v = 0; v < 8; ++v)
            out[(size_t)(m0 + hr + v) * FHID + j * 16 + lm] = c[j][v];
}

// ---------------- BatchNorm (biased var) + ReLU ----------------
__global__ void bn_stats(const float* __restrict__ h, int n,
                         float* __restrict__ sums, float* __restrict__ sumsq) {
    __shared__ float ls[FHID], lss[FHID];
    int tid = threadIdx.x;
    if (tid < FHID) { ls[tid] = 0.0f; lss[tid] = 0.0f; }
    __syncthreads();
    int f = tid & (FHID - 1);
    int rl = tid >> 6;                       // 0..3 row-lanes per block
    float s = 0.0f, ss = 0.0f;
    for (int r = blockIdx.x * 4 + rl; r < n; r += gridDim.x * 4) {
        float v = h[(size_t)r * FHID + f];
        s += v; ss += v * v;
    }
    atomicAdd(&ls[f], s);
    atomicAdd(&lss[f], ss);
    __syncthreads();
    if (tid < FHID) {
        atomicAdd(&sums[tid], ls[tid]);
        atomicAdd(&sumsq[tid], lss[tid]);
    }
}

__global__ void bn_finalize(const float* __restrict__ sums, const float* __restrict__ sumsq,
                            const float* __restrict__ g, const float* __restrict__ bt,
                            float n, float* __restrict__ scale, float* __restrict__ shift) {
    int f = threadIdx.x;
    if (f < FHID) {
        float m = sums[f] / n;
        float var = sumsq[f] / n - m * m;
        float inv = rsqrtf(var + BN_EPS);
        float sc = g[f] * inv;
        scale[f] = sc;
        shift[f] = bt[f] - m * sc;
    }
}

__global__ void bn_relu_apply(float* __restrict__ h, const float* __restrict__ scale,
                              const float* __restrict__ shift, int total) {
    int i = blockIdx.x * blockDim.x + threadIdx.x;
    if (i < total) {
        int f = i & (FHID - 1);
        h[i] = fmaxf(0.0f, fmaf(h[i], scale[f], shift[f]));
    }
}

// ---------------- final mix: out[n,:6] = [h1|h2p[c1]|h3p[c2]|x] @ Wm + bm ----------------
__global__ void mix_kernel(const float* __restrict__ h1, const float* __restrict__ h2,
                           const float* __restrict__ h3, const float* __restrict__ x,
                           const int* __restrict__ c1, const int* __restrict__ c2,
                           const float* __restrict__ Wm, const float* __restrict__ bm,
                           float* __restrict__ out, int n) {
    __shared__ float w[FMIX * FOUT + 8];
    for (int i = threadIdx.x; i < FMIX * FOUT; i += blockDim.x) w[i] = Wm[i];
    if (threadIdx.x < FOUT) w[FMIX * FOUT + threadIdx.x] = bm[threadIdx.x];
    __syncthreads();
    int node = blockIdx.x * blockDim.x + threadIdx.x;
    if (node >= n) return;

    float acc[FOUT];
    #pragma unroll
    for (int o = 0; o < FOUT; ++o) acc[o] = w[FMIX * FOUT + o];

    const float* r1 = h1 + (size_t)node * FHID;
    const float* r2 = h2 + (size_t)c1[node] * FHID;
    const float* r3 = h3 + (size_t)c2[node] * FHID;
    const float* rx = x + (size_t)node * FIN;

    for (int f = 0; f < FHID; ++f) {
        float v = r1[f];
        #pragma unroll
        for (int o = 0; o < FOUT; ++o) acc[o] = fmaf(v, w[f * FOUT + o], acc[o]);
    }
    for (int f = 0; f < FHID; ++f) {
        float v = r2[f];
        #pragma unroll
        for (int o = 0; o < FOUT; ++o) acc[o] = fmaf(v, w[(FHID + f) * FOUT + o], acc[o]);
    }
    for (int f = 0; f < FHID; ++f) {
        float v = r3[f];
        #pragma unroll
        for (int o = 0; o < FOUT; ++o) acc[o] = fmaf(v, w[(2 * FHID + f) * FOUT + o], acc[o]);
    }
    for (int f = 0; f < FIN; ++f) {
        float v = rx[f];
        #pragma unroll
        for (int o = 0; o < FOUT; ++o) acc[o] = fmaf(v, w[(3 * FHID + f) * FOUT + o], acc[o]);
    }
    #pragma unroll
    for (int o = 0; o < FOUT; ++o) out[(size_t)node * FOUT + o] = acc[o];
}

// =====================================================================
static inline int ceil_div(int a, int b) { return (a + b - 1) / b; }

extern "C" void kernel_launch(void* const* d_in, const int* in_sizes, int n_in,
                              void* d_out, int out_size, void* d_ws, size_t ws_size,
                              hipStream_t stream) {
    (void)in_sizes; (void)n_in; (void)out_size; (void)ws_size;
    const float* x   = (const float*)d_in[0];
    const int*   ei  = (const int*)d_in[1];   // [2, NE]
    const int*   cl1 = (const int*)d_in[2];
    const int*   cl2 = (const int*)d_in[3];
    const int*   e1  = (const int*)d_in[4];   // [2, NE1]
    const int*   e2  = (const int*)d_in[5];   // [2, NE2]
    const float* W1  = (const float*)d_in[6];
    const float* g1  = (const float*)d_in[8];
    const float* bt1 = (const float*)d_in[9];
    const float* W2  = (const float*)d_in[10];
    const float* g2  = (const float*)d_in[12];
    const float* bt2 = (const float*)d_in[13];
    const float* W3  = (const float*)d_in[14];
    const float* g3  = (const float*)d_in[16];
    const float* bt3 = (const float*)d_in[17];
    const float* Wm  = (const float*)d_in[18];
    const float* bm  = (const float*)d_in[19];
    float* out = (float*)d_out;

    // ---- workspace bump allocator (floats) ----
    float* ws = (float*)d_ws;
    size_t off = 0;
    auto alloc = [&](size_t nfl) {
        float* p = ws + off;
        off += (nfl + 63) & ~(size_t)63;
        return p;
    };
    float* disN   = alloc(NN);
    float* normN  = alloc(NE);
    float* disC1  = alloc(NC1);
    float* normC1 = alloc(NE1);
    float* disC2  = alloc(NC2);
    float* normC2 = alloc(NE2);
    float* cnt1   = alloc(NC1);
    float* cnt2   = alloc(NC2);
    float* xp1    = alloc((size_t)NC1 * FIN);
    float* xp2    = alloc((size_t)NC2 * FIN);
    float* TA     = alloc((size_t)NN * FIN);
    float* TB     = alloc((size_t)NN * FIN);
    float* TC     = alloc((size_t)NN * FIN);
    float* h1     = alloc((size_t)NN * FHID);
    float* h2p    = alloc((size_t)NC1 * FHID);
    float* h3p    = alloc((size_t)NC2 * FHID);
    float* sums   = alloc(FHID);
    float* sumsq  = alloc(FHID);
    float* scale  = alloc(FHID);
    float* shift  = alloc(FHID);

    const int TB_ = 256;

    // ---- graph normalization coefficients (deg -> dis in place -> edge norm) ----
    auto make_norm = [&](const int* edges, int E, float* dis, float* norm, int n) {
        hipMemsetAsync(dis, 0, (size_t)n * 4, stream);
        deg_kernel<<<ceil_div(E, TB_), TB_, 0, stream>>>(edges + E, dis, E);   // dst row
        dis_kernel<<<ceil_div(n, TB_), TB_, 0, stream>>>(dis, n);
        norm_kernel<<<ceil_div(E, TB_), TB_, 0, stream>>>(edges, edges + E, dis, norm, E);
    };
    make_norm(ei, NE, disN, normN, NN);
    make_norm(e1, NE1, disC1, normC1, NC1);
    make_norm(e2, NE2, disC2, normC2, NC2);

    // ---- cluster pooling (segment mean of x) ----
    hipMemsetAsync(cnt1, 0, (size_t)NC1 * 4, stream);
    hipMemsetAsync(xp1, 0, (size_t)NC1 * FIN * 4, stream);
    pool_scatter<<<ceil_div(NN, TB_), TB_, 0, stream>>>(x, cl1, xp1, cnt1, NN);
    pool_div<<<ceil_div(NC1 * FIN, TB_), TB_, 0, stream>>>(xp1, cnt1, NC1);

    hipMemsetAsync(cnt2, 0, (size_t)NC2 * 4, stream);
    hipMemsetAsync(xp2, 0, (size_t)NC2 * FIN * 4, stream);
    pool_scatter<<<ceil_div(NN, TB_), TB_, 0, stream>>>(x, cl2, xp2, cnt2, NN);
    pool_div<<<ceil_div(NC2 * FIN, TB_), TB_, 0, stream>>>(xp2, cnt2, NC2);

    // ---- one ChebConv(K=6) + BN + ReLU branch ----
    // NOTE: the conv bias b is a per-feature constant that BatchNorm's mean
    // subtraction cancels exactly, so it is omitted.
    auto run_branch = [&](const float* xin, int n, const int* edges, const float* norm, int E,
                          const float* W, const float* g, const float* bt, float* h) {
        int ntiles = n / 16;                       // n is always a multiple of 16 here
        int wblocks = ceil_div(ntiles, TB_ / 32);  // 8 waves / block
        hipMemsetAsync(h, 0, (size_t)n * FHID * 4, stream);

        // k = 0: h += Tx0 @ W0
        wmma_acc_kernel<<<wblocks, TB_, 0, stream>>>(xin, W, h, ntiles);

        // k = 1: Tx1 = prop(Tx0); h += Tx1 @ W1
        hipMemsetAsync(TA, 0, (size_t)n * FIN * 4, stream);
        scatter_prop<<<ceil_div(E, TB_), TB_, 0, stream>>>(xin, edges, edges + E, norm, 1.0f, TA, E);
        wmma_acc_kernel<<<wblocks, TB_, 0, stream>>>(TA, W + 1 * FIN * FHID, h, ntiles);

        // k = 2..5: Tx2 = 2*prop(Tx1) - Tx0 ; h += Tx2 @ Wk
        float* bufs[3] = {TA, TB, TC};
        const float* c0 = xin;
        float* c1 = TA;
        int nextIdx = 1;
        for (int k = 2; k < KCH; ++k) {
            float* c2 = bufs[nextIdx];
            nextIdx = (nextIdx + 1) % 3;
            neg_copy<<<ceil_div(n * FIN, TB_), TB_, 0, stream>>>(c0, c2, n * FIN);
            scatter_prop<<<ceil_div(E, TB_), TB_, 0, stream>>>(c1, edges, edges + E, norm, 2.0f, c2, E);
            wmma_acc_kernel<<<wblocks, TB_, 0, stream>>>(c2, W + k * FIN * FHID, h, ntiles);
            c0 = c1;
            c1 = c2;
        }

        // BatchNorm (biased var, as torch BN forward) + ReLU, fused apply
        hipMemsetAsync(sums, 0, FHID * 4, stream);
        hipMemsetAsync(sumsq, 0, FHID * 4, stream);
        bn_stats<<<256, 256, 0, stream>>>(h, n, sums, sumsq);
        bn_finalize<<<1, 64, 0, stream>>>(sums, sumsq, g, bt, (float)n, scale, shift);
        bn_relu_apply<<<ceil_div(n * FHID, TB_), TB_, 0, stream>>>(h, scale, shift, n * FHID);
    };

    run_branch(x,   NN,  ei, normN,  NE,  W1, g1, bt1, h1);
    run_branch(xp1, NC1, e1, normC1, NE1, W2, g2, bt2, h2p);
    run_branch(xp2, NC2, e2, normC2, NE2, W3, g3, bt3, h3p);

    // ---- mix (K=1 Cheb == dense 216->6 matmul over gathered concat) ----
    mix_kernel<<<ceil_div(NN, TB_), TB_, 0, stream>>>(h1, h2p, h3p, x, cl1, cl2, Wm, bm, out, NN);
}